// SubbandFilter_2293512536591
// MI455X (gfx1250) — compile-verified
//
#include <hip/hip_runtime.h>
#include <hip/hip_bf16.h>

#define LEN     262144
#define HALF    (LEN / 2)
#define BATCH   8          // B*C = 4*2 signals
#define NBANDS  10
#define TAPS    39
#define FACTOR  10
#define LDOWN   26215      // decimated samples: indices 0,10,...,262140
#define TILES   1639       // ceil(26215 / 16)
#define PI_F    3.14159265358979323846f

// padded analytic-signal row: [front pad | LEN | back pad]
#define XPAD_FRONT 32
#define XPAD_BACK  128
#define XSTRIDE    (XPAD_FRONT + LEN + XPAD_BACK)   // 262304

typedef float v2f __attribute__((ext_vector_type(2)));
typedef float v8f __attribute__((ext_vector_type(8)));

#if defined(__has_builtin)
#if __has_builtin(__builtin_amdgcn_wmma_f32_16x16x4_f32)
#define HAVE_WMMA_F32X4 1
#endif
#endif

// ---------------- Stage 0: real -> complex load ----------------
__global__ void sbf_load_real(const float* __restrict__ x, float2* __restrict__ A) {
    size_t i = (size_t)blockIdx.x * blockDim.x + threadIdx.x;
    if (i < (size_t)BATCH * LEN) A[i] = make_float2(x[i], 0.0f);
}

// ---------------- Stockham radix-2 pass (fwd: sgn=+1, inv: sgn=-1) ----------------
// Output row layout parameterized so the final pass can emit the padded xa buffer.
__global__ void sbf_fft_pass(const float2* __restrict__ in, float2* __restrict__ out,
                             int s, float sgn, long ostride, int ooff) {
    int j = blockIdx.x * blockDim.x + threadIdx.x;   // [0, HALF)
    int b = blockIdx.y;
    const float2* X = in  + (size_t)b * LEN;
    float2*       Y = out + (size_t)b * ostride + ooff;

    int Ls = 1 << s;
    int q  = j >> s;
    int r  = j & (Ls - 1);

    float2 a  = X[j];
    float2 bb = X[j + HALF];

    float ang = sgn * (-PI_F) * ((float)r / (float)Ls);
    float c, sn;
    sincosf(ang, &c, &sn);
    float2 wb = make_float2(bb.x * c - bb.y * sn, bb.x * sn + bb.y * c);

    int o = (q << (s + 1)) + r;
    Y[o]      = make_float2(a.x + wb.x, a.y + wb.y);
    Y[o + Ls] = make_float2(a.x - wb.x, a.y - wb.y);
}

// ---------------- Hilbert mask (+ fold 1/L of the inverse FFT) ----------------
__global__ void sbf_hilbert_mask(float2* __restrict__ A) {
    int k = blockIdx.x * blockDim.x + threadIdx.x;
    int b = blockIdx.y;
    float h = (k == 0 || k == HALF) ? 1.0f : (k < HALF ? 2.0f : 0.0f);
    float scale = h * (1.0f / (float)LEN);
    size_t idx = (size_t)b * LEN + k;
    float2 v = A[idx];
    A[idx] = make_float2(v.x * scale, v.y * scale);
}

// ---------------- zero the pads of the padded xa buffer ----------------
__global__ void sbf_zero_pads(float2* __restrict__ xp) {
    int b = blockIdx.x;
    int t = threadIdx.x;                     // 0..159 (32 front + 128 back)
    float2* row = xp + (size_t)b * XSTRIDE;
    if (t < XPAD_FRONT) row[t] = make_float2(0.0f, 0.0f);
    else                row[XPAD_FRONT + LEN + (t - XPAD_FRONT)] = make_float2(0.0f, 0.0f);
}

// ---------------- FIR bank (WMMA f32) + demod + decimate ----------------
// One wave per 16-sample output tile. LDS-staged, guard-free inner loop:
// C[m, n] = sum_t W[m, t] * xa[10*(d0+n) + 19 - t]  (real & imag accumulators)
__global__ void __launch_bounds__(32)
sbf_fir_demod(const float2* __restrict__ xp, const float* __restrict__ w,
              const float* __restrict__ freq, float* __restrict__ out) {
    __shared__ float2 sX[192];       // xa window [10*d0-20, 10*d0+169]
    __shared__ float  sW[16 * 40];   // zero-padded filter bank

    const int lane = threadIdx.x;    // 0..31
    const int bc   = blockIdx.y;     // signal index (b*2 + c)
    const int d0   = blockIdx.x * 16;
    const float2* X = xp + (size_t)bc * XSTRIDE + XPAD_FRONT;  // X[i] valid for i in [-32, LEN+128)

    // stage padded W (guards resolved once, outside the hot loop)
    for (int i = lane; i < 16 * 40; i += 32) {
        int m = i / 40, t = i - m * 40;
        sW[i] = (m < NBANDS && t < TAPS) ? w[m * TAPS + t] : 0.0f;
    }
    // stage contiguous xa window (coalesced b64 loads; pads make this guard-free)
    const long w0 = (long)FACTOR * d0 - 20;
    for (int i = lane; i < 190; i += 32) sX[i] = X[w0 + i];
    __syncthreads();

    const int n  = lane & 15;        // A-row (band) / B-col (sample) for this lane
    const int kg = lane >> 4;        // K half-group: lanes 16-31 hold K+2,K+3

#ifdef HAVE_WMMA_F32X4
    v8f cr = {};
    v8f ci = {};
#pragma unroll
    for (int k0 = 0; k0 < 40; k0 += 4) {     // taps padded 39 -> 40
        const int kA = k0 + kg * 2;          // even -> 8B-aligned LDS reads
        v2f a = *(const v2f*)(sW + n * 40 + kA);
        float2 x0 = sX[10 * n + 39 - kA];    // local = global - (10*d0 - 20)
        float2 x1 = sX[10 * n + 38 - kA];
        v2f br; br.x = x0.x; br.y = x1.x;
        v2f bi; bi.x = x0.y; bi.y = x1.y;
        cr = __builtin_amdgcn_wmma_f32_16x16x4_f32(false, a, false, br, (short)0, cr, false, false);
        ci = __builtin_amdgcn_wmma_f32_16x16x4_f32(false, a, false, bi, (short)0, ci, false, false);
    }

    // C layout: reg r holds (M = r + 8*kg, N = n)
    const int d = d0 + n;
#pragma unroll
    for (int rI = 0; rI < 8; ++rI) {
        int m = rI + 8 * kg;
        if (m < NBANDS && d < LDOWN) {
            float sr = cr[rI], si = ci[rI];
            float omega = freq[m] * (PI_F / 22050.0f);
            float ph = omega * (float)(FACTOR * d);
            float cth, sth;
            sincosf(ph, &cth, &sth);
            size_t o = ((size_t)bc * NBANDS + m) * LDOWN + d;
            out[o] = sr * cth - si * sth;                                   // real plane
            out[(size_t)BATCH * NBANDS * LDOWN + o] = sr * sth + si * cth;  // imag plane
        }
    }
#else
    // LDS-based scalar fallback (keeps compile green if the builtin is absent)
    if (lane < 16) {
        int d = d0 + lane;
        if (d < LDOWN) {
            for (int m = 0; m < NBANDS; ++m) {
                float sr = 0.0f, si = 0.0f;
                for (int t = 0; t < TAPS; ++t) {
                    float2 v = sX[10 * lane + 39 - t];
                    float wt = sW[m * 40 + t];
                    sr += wt * v.x;
                    si += wt * v.y;
                }
                float omega = freq[m] * (PI_F / 22050.0f);
                float ph = omega * (float)(FACTOR * d);
                float cth, sth;
                sincosf(ph, &cth, &sth);
                size_t o = ((size_t)bc * NBANDS + m) * LDOWN + d;
                out[o] = sr * cth - si * sth;
                out[(size_t)BATCH * NBANDS * LDOWN + o] = sr * sth + si * cth;
            }
        }
    }
#endif
}

extern "C" void kernel_launch(void* const* d_in, const int* in_sizes, int n_in,
                              void* d_out, int out_size, void* d_ws, size_t ws_size,
                              hipStream_t stream) {
    (void)in_sizes; (void)n_in; (void)out_size; (void)ws_size;
    const float* x    = (const float*)d_in[0];
    const float* w    = (const float*)d_in[1];
    const float* freq = (const float*)d_in[2];
    float* out = (float*)d_out;

    // workspace: two complex ping-pong buffers (16 MB each) + padded xa (~16.8 MB)
    float2* A  = (float2*)d_ws;
    float2* Bb = A  + (size_t)BATCH * LEN;
    float2* XP = Bb + (size_t)BATCH * LEN;

    // 1) real -> complex
    sbf_load_real<<<dim3((BATCH * LEN) / 256), 256, 0, stream>>>(x, A);

    // 2) forward FFT: 18 Stockham radix-2 passes (even count -> back in A)
    float2 *cur = A, *nxt = Bb;
    for (int s = 0; s < 18; ++s) {
        sbf_fft_pass<<<dim3(HALF / 256, BATCH), 256, 0, stream>>>(cur, nxt, s, 1.0f, (long)LEN, 0);
        float2* t = cur; cur = nxt; nxt = t;
    }

    // 3) Hilbert mask (analytic-signal spectrum, 1/L folded in)
    sbf_hilbert_mask<<<dim3(LEN / 256, BATCH), 256, 0, stream>>>(cur);

    // 4) inverse FFT: 17 normal conjugate passes, final pass emits padded layout
    for (int s = 0; s < 17; ++s) {
        sbf_fft_pass<<<dim3(HALF / 256, BATCH), 256, 0, stream>>>(cur, nxt, s, -1.0f, (long)LEN, 0);
        float2* t = cur; cur = nxt; nxt = t;
    }
    sbf_fft_pass<<<dim3(HALF / 256, BATCH), 256, 0, stream>>>(cur, XP, 17, -1.0f,
                                                              (long)XSTRIDE, XPAD_FRONT);
    sbf_zero_pads<<<dim3(BATCH), 160, 0, stream>>>(XP);

    // 5) FIR bank as f32 WMMA GEMM at decimated positions + demod + store
    sbf_fir_demod<<<dim3(TILES, BATCH), 32, 0, stream>>>(XP, w, freq, out);
}